// LinearGraph_27951647163108
// MI455X (gfx1250) — compile-verified
//
#include <hip/hip_runtime.h>
#include <hip/hip_bf16.h>

#define N_NODES  50000
#define N_EDGES  800000
#define HID      256
#define NGRAPH   512
#define NCLASS   10

typedef float v2f __attribute__((ext_vector_type(2)));
typedef float v8f __attribute__((ext_vector_type(8)));

__device__ __forceinline__ void atomAddF(float* p, float v) {
    __hip_atomic_fetch_add(p, v, __ATOMIC_RELAXED, __HIP_MEMORY_SCOPE_AGENT);
}

// ---------------------------------------------------------------- utilities
__global__ void fill_kernel(float* __restrict__ p, float v, int n) {
    int i = blockIdx.x * blockDim.x + threadIdx.x;
    if (i < n) p[i] = v;
}

// deg[col[e]] += 1  (deg pre-initialized to 1.0 for the self loop)
__global__ void deg_kernel(const int* __restrict__ col, float* __restrict__ deg, int e) {
    int i = blockIdx.x * blockDim.x + threadIdx.x;
    if (i < e) atomAddF(&deg[col[i]], 1.0f);
}

// dinv = rsqrt(deg), in place (deg >= 1 always due to self loops)
__global__ void dinv_kernel(float* __restrict__ d, int n) {
    int i = blockIdx.x * blockDim.x + threadIdx.x;
    if (i < n) d[i] = rsqrtf(d[i]);
}

// ------------------------------------------------- fp32 WMMA GEMM  Out = X@W (+bias)
// X: [M,256] row-major, W: [256,256] row-major, Out: [M,256].
// One wave per 16x64 output strip; 4 WMMA accumulators; K-loop of 64 x (16x16x4).
__global__ __launch_bounds__(256) void gemm_wmma(const float* __restrict__ X,
                                                 const float* __restrict__ W,
                                                 const float* __restrict__ bias,
                                                 float* __restrict__ Out, int M) {
    const int lane  = threadIdx.x & 31;
    const int wid   = (blockIdx.x * blockDim.x + threadIdx.x) >> 5;   // global wave id
    const int tileM = wid >> 2;                                       // 16-row tile
    const int cblk  = wid & 3;                                        // 64-col block
    if (tileM * 16 >= M) return;                                      // wave-uniform exit

    const int row0 = tileM * 16;
    const int col0 = cblk * 64;
    const int n    = lane & 15;            // A-row / B-col / D-col index for this lane
    const int kh   = (lane >> 4) << 1;     // 0 for lanes 0-15, 2 for lanes 16-31

    const float* __restrict__ Arow = X + (size_t)(row0 + n) * HID;

    v8f acc0 = {}, acc1 = {}, acc2 = {}, acc3 = {};

    for (int k0 = 0; k0 < HID; k0 += 4) {
        // A fragment: 16x4 f32. VGPR0 = K=(k0+kh), VGPR1 = K=(k0+kh+1)
        v2f a;
        a.x = Arow[k0 + kh];
        a.y = Arow[k0 + kh + 1];

        // B fragments: 4x16 f32, rows striped across VGPRs like C/D.
        const float* __restrict__ Wr = W + (size_t)(k0 + kh) * HID + col0 + n;
        v2f b0, b1, b2, b3;
        b0.x = Wr[0];       b0.y = Wr[HID];
        b1.x = Wr[16];      b1.y = Wr[HID + 16];
        b2.x = Wr[32];      b2.y = Wr[HID + 32];
        b3.x = Wr[48];      b3.y = Wr[HID + 48];

        acc0 = __builtin_amdgcn_wmma_f32_16x16x4_f32(false, a, false, b0, (short)0, acc0, false, false);
        acc1 = __builtin_amdgcn_wmma_f32_16x16x4_f32(false, a, false, b1, (short)0, acc1, false, false);
        acc2 = __builtin_amdgcn_wmma_f32_16x16x4_f32(false, a, false, b2, (short)0, acc2, false, false);
        acc3 = __builtin_amdgcn_wmma_f32_16x16x4_f32(false, a, false, b3, (short)0, acc3, false, false);
    }

    float bv0 = 0.f, bv1 = 0.f, bv2 = 0.f, bv3 = 0.f;
    if (bias) {
        bv0 = bias[col0 + n];
        bv1 = bias[col0 + 16 + n];
        bv2 = bias[col0 + 32 + n];
        bv3 = bias[col0 + 48 + n];
    }

    // D layout: VGPR r -> row r (lanes 0-15) / row r+8 (lanes 16-31), col = lane&15
    const int roff = (lane >> 4) * 8;
    for (int r = 0; r < 8; ++r) {
        float* __restrict__ Orow = Out + (size_t)(row0 + roff + r) * HID + col0 + n;
        Orow[0]  = acc0[r] + bv0;
        Orow[16] = acc1[r] + bv1;
        Orow[32] = acc2[r] + bv2;
        Orow[48] = acc3[r] + bv3;
    }
}

// ----------------------------------------- edge gather + normalize + scatter-add
// One wave per edge (incl. N self-loops appended past E). Agg[col] += H2[row]*norm.
__global__ __launch_bounds__(256) void scatter_kernel(const float* __restrict__ H2,
                                                      const float* __restrict__ dinv,
                                                      const int* __restrict__ src,
                                                      const int* __restrict__ dst,
                                                      float* __restrict__ Agg) {
    const int lane = threadIdx.x & 31;
    const int e    = (blockIdx.x * blockDim.x + threadIdx.x) >> 5;
    if (e >= N_EDGES + N_NODES) return;

    int r, c;
    if (e < N_EDGES) { r = src[e]; c = dst[e]; }
    else             { r = e - N_EDGES; c = r; }

    const float w = dinv[r] * dinv[c];

    const float4* __restrict__ S = (const float4*)(H2 + (size_t)r * HID);
    float* __restrict__ D = Agg + (size_t)c * HID + lane * 4;

    float4 v0 = S[lane];        // features [4*lane .. 4*lane+3]
    float4 v1 = S[lane + 32];   // features [128+4*lane .. ]

    atomAddF(D + 0, v0.x * w);
    atomAddF(D + 1, v0.y * w);
    atomAddF(D + 2, v0.z * w);
    atomAddF(D + 3, v0.w * w);
    atomAddF(D + 128 + 0, v1.x * w);
    atomAddF(D + 128 + 1, v1.y * w);
    atomAddF(D + 128 + 2, v1.z * w);
    atomAddF(D + 128 + 3, v1.w * w);
}

// h = relu(h + bias), in place
__global__ void bias_relu_kernel(float* __restrict__ h, const float* __restrict__ b, int total) {
    int i = blockIdx.x * blockDim.x + threadIdx.x;
    if (i < total) h[i] = fmaxf(h[i] + b[i & (HID - 1)], 0.0f);
}

// pooled sums: psum[batch[n]][f] += h[n][f]
__global__ void pool_sum_kernel(const float* __restrict__ h, const int* __restrict__ batch,
                                float* __restrict__ psum) {
    int i = blockIdx.x * blockDim.x + threadIdx.x;
    if (i >= N_NODES * HID) return;
    int node = i >> 8;
    int f    = i & (HID - 1);
    atomAddF(&psum[(size_t)batch[node] * HID + f], h[i]);
}

__global__ void pool_cnt_kernel(const int* __restrict__ batch, float* __restrict__ pcnt) {
    int i = blockIdx.x * blockDim.x + threadIdx.x;
    if (i < N_NODES) atomAddF(&pcnt[batch[i]], 1.0f);
}

// out[g][c] = (1/max(cnt,1)) * dot(psum[g], outW[:,c]) + outB[c]
__global__ void final_kernel(const float* __restrict__ psum, const float* __restrict__ pcnt,
                             const float* __restrict__ oW, const float* __restrict__ ob,
                             float* __restrict__ out) {
    int i = blockIdx.x * blockDim.x + threadIdx.x;
    if (i >= NGRAPH * NCLASS) return;
    int g = i / NCLASS;
    int c = i % NCLASS;
    float inv = 1.0f / fmaxf(pcnt[g], 1.0f);
    const float* __restrict__ s = psum + (size_t)g * HID;
    float acc = 0.0f;
    for (int f = 0; f < HID; ++f) acc += s[f] * oW[f * NCLASS + c];
    out[i] = acc * inv + ob[c];
}

// ---------------------------------------------------------------------------
extern "C" void kernel_launch(void* const* d_in, const int* in_sizes, int n_in,
                              void* d_out, int out_size, void* d_ws, size_t ws_size,
                              hipStream_t stream) {
    (void)in_sizes; (void)n_in; (void)out_size; (void)ws_size;

    const float* x     = (const float*)d_in[0];
    const int*   ei    = (const int*)d_in[1];     // [2, E] flat: row then col
    const int*   batch = (const int*)d_in[2];
    const float* encW  = (const float*)d_in[3];
    const float* encB  = (const float*)d_in[4];
    const float* convW = (const float*)d_in[5];   // [2,256,256]
    const float* convB = (const float*)d_in[6];   // [2,256]
    const float* outW  = (const float*)d_in[7];
    const float* outB  = (const float*)d_in[8];
    float* out = (float*)d_out;

    const int* row = ei;
    const int* col = ei + N_EDGES;

    // workspace carve-out
    char* ws = (char*)d_ws;
    size_t off = 0;
    auto carve = [&](size_t bytes) {
        void* p = ws + off;
        off += (bytes + 255) & ~(size_t)255;
        return p;
    };
    float* dinv = (float*)carve((size_t)N_NODES * 4);
    float* bufA = (float*)carve((size_t)N_NODES * HID * 4);
    float* bufB = (float*)carve((size_t)N_NODES * HID * 4);
    float* psum = (float*)carve((size_t)NGRAPH * HID * 4);
    float* pcnt = (float*)carve((size_t)NGRAPH * 4);

    const int totalNH   = N_NODES * HID;
    const int gNH       = (totalNH + 255) / 256;
    const int gN        = (N_NODES + 255) / 256;
    const int gE        = (N_EDGES + 255) / 256;
    const int gemmWaves = (N_NODES / 16) * 4;                       // 12500
    const int gGemm     = (gemmWaves * 32 + 255) / 256;             // 1563
    const int gScat     = ((N_EDGES + N_NODES) * 32 + 255) / 256;   // 106250

    // --- degree normalization ---
    fill_kernel<<<gN, 256, 0, stream>>>(dinv, 1.0f, N_NODES);             // self loop
    deg_kernel<<<gE, 256, 0, stream>>>(col, dinv, N_EDGES);
    dinv_kernel<<<gN, 256, 0, stream>>>(dinv, N_NODES);

    // --- encoder: bufA = x @ encW + encB ---
    gemm_wmma<<<gGemm, 256, 0, stream>>>(x, encW, encB, bufA, N_NODES);

    // --- 2 GCN layers ---
    for (int l = 0; l < 2; ++l) {
        const float* W = convW + (size_t)l * HID * HID;
        const float* b = convB + (size_t)l * HID;
        // h2 = h @ W
        gemm_wmma<<<gGemm, 256, 0, stream>>>(bufA, W, nullptr, bufB, N_NODES);
        // agg = 0 (reuse bufA: h is dead after the gemm)
        fill_kernel<<<gNH, 256, 0, stream>>>(bufA, 0.0f, totalNH);
        // agg[col] += h2[row] * dinv[row]*dinv[col]
        scatter_kernel<<<gScat, 256, 0, stream>>>(bufB, dinv, row, col, bufA);
        // h = relu(agg + b)
        bias_relu_kernel<<<gNH, 256, 0, stream>>>(bufA, b, totalNH);
    }

    // --- global mean pool + classifier ---
    fill_kernel<<<(NGRAPH * HID + 255) / 256, 256, 0, stream>>>(psum, 0.0f, NGRAPH * HID);
    fill_kernel<<<(NGRAPH + 255) / 256, 256, 0, stream>>>(pcnt, 0.0f, NGRAPH);
    pool_sum_kernel<<<gNH, 256, 0, stream>>>(bufA, batch, psum);
    pool_cnt_kernel<<<gN, 256, 0, stream>>>(batch, pcnt);
    final_kernel<<<(NGRAPH * NCLASS + 255) / 256, 256, 0, stream>>>(psum, pcnt, outW, outB, out);
}